// ShapeClassifierViz_51196010168909
// MI455X (gfx1250) — compile-verified
//
#include <hip/hip_runtime.h>
#include <hip/hip_bf16.h>

typedef float v2f __attribute__((ext_vector_type(2)));
typedef float v8f __attribute__((ext_vector_type(8)));

// Non-returning fp32 atomic add, device scope -> global_atomic_add_f32 (STOREcnt).
__device__ __forceinline__ void atomic_add_f32(float* p, float v) {
    __hip_atomic_fetch_add(p, v, __ATOMIC_RELAXED, __HIP_MEMORY_SCOPE_AGENT);
}

__global__ void fill_f32(float* __restrict__ p, float v, int n) {
    int i = blockIdx.x * blockDim.x + threadIdx.x;
    if (i < n) p[i] = v;
}

// deg[dst[e]] += 1  (self-loop's +1 is pre-initialized by fill_f32(deg, 1.0f))
__global__ void deg_accum(const int* __restrict__ dst, float* __restrict__ deg, int E) {
    int e = blockIdx.x * blockDim.x + threadIdx.x;
    if (e < E) atomic_add_f32(&deg[dst[e]], 1.0f);
}

__global__ void rsqrt_inplace(float* __restrict__ d, int n) {
    int i = blockIdx.x * blockDim.x + threadIdx.x;
    if (i < n) d[i] = rsqrtf(d[i]);   // deg >= 1 always (self loop)
}

// C[nrows x 16] = A[nrows x K] @ B[K x 16] via V_WMMA_F32_16X16X4_F32.
// One wave per 16-row tile; K accumulated in steps of 4 (zero-padded via
// clamped-address loads + value selects -> no exec-mask branches).
template <int K>
__global__ void gcn_gemm_wmma(const float* __restrict__ A, const float* __restrict__ B,
                              float* __restrict__ C, int nrows) {
    const int lane  = threadIdx.x & 31;
    const int wave  = blockIdx.x * (blockDim.x >> 5) + (threadIdx.x >> 5);
    const int ntile = (nrows + 15) >> 4;
    if (wave >= ntile) return;             // uniform per-wave exit: EXEC all-1s at WMMA

    const int row0  = wave << 4;
    const int m     = lane & 15;           // A: row-in-tile ; B/C: column
    const int khalf = (lane >> 4) << 1;    // lanes 0-15 -> K pair {0,1}; 16-31 -> {2,3}
    constexpr int KSTEPS = (K + 3) / 4;

    // Stage B (weights) in registers. Loads always in-bounds (clamped index),
    // padding handled by value select (v_cndmask, no branch).
    v2f breg[KSTEPS];
#pragma unroll
    for (int s = 0; s < KSTEPS; ++s) {
        const int kb = s * 4 + khalf;
        if constexpr ((K & 3) == 0) {
            breg[s].x = B[kb * 16 + m];
            breg[s].y = B[(kb + 1) * 16 + m];
        } else {
            const int k0 = (kb     < K) ? kb     : K - 1;
            const int k1 = (kb + 1 < K) ? kb + 1 : K - 1;
            const float bv0 = B[k0 * 16 + m];
            const float bv1 = B[k1 * 16 + m];
            breg[s].x = (kb     < K) ? bv0 : 0.0f;
            breg[s].y = (kb + 1 < K) ? bv1 : 0.0f;
        }
    }

    v8f c = {};
    const int row   = row0 + m;
    const int rbase = row0 + ((lane >> 4) << 3);

    if (row0 + 16 <= nrows) {
        // Fast path (always taken for nrows % 16 == 0): unconditional loads/stores.
#pragma unroll
        for (int s = 0; s < KSTEPS; ++s) {
            const int kb = s * 4 + khalf;
            v2f a;
            if constexpr ((K & 3) == 0) {
                a = *(const v2f*)(A + (size_t)row * K + kb);     // b64 load, 8B-aligned
            } else {
                const int k0 = (kb     < K) ? kb     : K - 1;
                const int k1 = (kb + 1 < K) ? kb + 1 : K - 1;
                const float a0 = A[(size_t)row * K + k0];
                const float a1 = A[(size_t)row * K + k1];
                a.x = (kb     < K) ? a0 : 0.0f;
                a.y = (kb + 1 < K) ? a1 : 0.0f;
            }
            c = __builtin_amdgcn_wmma_f32_16x16x4_f32(false, a, false, breg[s],
                                                      (short)0, c, false, false);
        }
#pragma unroll
        for (int j = 0; j < 8; ++j)
            C[(size_t)(rbase + j) * 16 + m] = c[j];   // lanes 0-15 / 16-31: 64B rows
    } else {
        // Ragged last tile: clamp row, mask values, guard stores.
        const int   rr    = (row < nrows) ? row : nrows - 1;
        const bool  rok   = row < nrows;
#pragma unroll
        for (int s = 0; s < KSTEPS; ++s) {
            const int kb = s * 4 + khalf;
            const int k0 = (kb     < K) ? kb     : K - 1;
            const int k1 = (kb + 1 < K) ? kb + 1 : K - 1;
            const float a0 = A[(size_t)rr * K + k0];
            const float a1 = A[(size_t)rr * K + k1];
            v2f a;
            a.x = (rok && kb     < K) ? a0 : 0.0f;
            a.y = (rok && kb + 1 < K) ? a1 : 0.0f;
            c = __builtin_amdgcn_wmma_f32_16x16x4_f32(false, a, false, breg[s],
                                                      (short)0, c, false, false);
        }
#pragma unroll
        for (int j = 0; j < 8; ++j) {
            const int r = rbase + j;
            if (r < nrows) C[(size_t)r * 16 + m] = c[j];
        }
    }
}

// agg[dst] += h[src] * (dinv[src]*dinv[dst])   — one thread per edge, b128 gathers,
// 16 non-returning f32 atomics. h/agg are L2-resident (6.4 MB each on 192 MB L2).
__global__ void edge_scatter(const float* __restrict__ h, const int* __restrict__ src,
                             const int* __restrict__ dst, const float* __restrict__ dinv,
                             float* __restrict__ agg, int E) {
    int e = blockIdx.x * blockDim.x + threadIdx.x;
    if (e >= E) return;
    const int s = src[e];
    const int d = dst[e];
    const float w = dinv[s] * dinv[d];
    const float4* hp = (const float4*)(h + (size_t)s * 16);
    float* ap = agg + (size_t)d * 16;
#pragma unroll
    for (int q = 0; q < 4; ++q) {
        float4 v = hp[q];
        atomic_add_f32(ap + q * 4 + 0, v.x * w);
        atomic_add_f32(ap + q * 4 + 1, v.y * w);
        atomic_add_f32(ap + q * 4 + 2, v.z * w);
        atomic_add_f32(ap + q * 4 + 3, v.w * w);
    }
}

// agg[i] = relu(agg[i] + h[i]*dinv[i]^2 + bias)   (self-loop folded in, in place)
__global__ void bias_relu_self(float* __restrict__ agg, const float* __restrict__ h,
                               const float* __restrict__ dinv,
                               const float* __restrict__ bias, int n) {
    int i = blockIdx.x * blockDim.x + threadIdx.x;
    if (i >= n) return;
    const float di = dinv[i];
    const float w  = di * di;
    const float4* hp = (const float4*)(h + (size_t)i * 16);
    float4*       ap = (float4*)(agg + (size_t)i * 16);
    const float4* bp = (const float4*)bias;
#pragma unroll
    for (int q = 0; q < 4; ++q) {
        float4 a = ap[q], hv = hp[q], b = bp[q];
        a.x = fmaxf(a.x + hv.x * w + b.x, 0.0f);
        a.y = fmaxf(a.y + hv.y * w + b.y, 0.0f);
        a.z = fmaxf(a.z + hv.z * w + b.z, 0.0f);
        a.w = fmaxf(a.w + hv.w * w + b.w, 0.0f);
        ap[q] = a;
    }
}

// Layer-2 epilogue fused with mean-pool accumulation:
// h2p = relu(agg + h*dinv^2 + b2); pool[batch[i]] += h2p; cnt[batch[i]] += 1
__global__ void relu_self_pool(const float* __restrict__ agg, const float* __restrict__ h,
                               const float* __restrict__ dinv, const float* __restrict__ bias,
                               const int* __restrict__ batch, float* __restrict__ pool,
                               float* __restrict__ cnt, int n) {
    int i = blockIdx.x * blockDim.x + threadIdx.x;
    if (i >= n) return;
    const float di = dinv[i];
    const float w  = di * di;
    const int   g  = batch[i];
    const float4* hp = (const float4*)(h + (size_t)i * 16);
    const float4* ap = (const float4*)(agg + (size_t)i * 16);
    const float4* bp = (const float4*)bias;
    float* ps = pool + (size_t)g * 16;
#pragma unroll
    for (int q = 0; q < 4; ++q) {
        float4 a = ap[q], hv = hp[q], b = bp[q];
        atomic_add_f32(ps + q * 4 + 0, fmaxf(a.x + hv.x * w + b.x, 0.0f));
        atomic_add_f32(ps + q * 4 + 1, fmaxf(a.y + hv.y * w + b.y, 0.0f));
        atomic_add_f32(ps + q * 4 + 2, fmaxf(a.z + hv.z * w + b.z, 0.0f));
        atomic_add_f32(ps + q * 4 + 3, fmaxf(a.w + hv.w * w + b.w, 0.0f));
    }
    atomic_add_f32(cnt + g, 1.0f);
}

__global__ void finalize_mean(const float* __restrict__ pool, const float* __restrict__ cnt,
                              float* __restrict__ out, int g16) {
    int i = blockIdx.x * blockDim.x + threadIdx.x;
    if (i < g16) out[i] = pool[i] / fmaxf(cnt[i >> 4], 1.0f);
}

extern "C" void kernel_launch(void* const* d_in, const int* in_sizes, int n_in,
                              void* d_out, int out_size, void* d_ws, size_t ws_size,
                              hipStream_t stream) {
    (void)n_in; (void)ws_size;
    const float* x     = (const float*)d_in[0];   // [N,5]
    const int*   ei    = (const int*)d_in[1];     // [2,E]
    const int*   batch = (const int*)d_in[2];     // [N]
    const float* W1    = (const float*)d_in[3];   // [5,16]
    const float* b1    = (const float*)d_in[4];   // [16]
    const float* W2    = (const float*)d_in[5];   // [16,16]
    const float* b2    = (const float*)d_in[6];   // [16]
    float*       out   = (float*)d_out;           // [G,16]

    const int N = in_sizes[2];
    const int E = in_sizes[1] / 2;
    const int G = out_size / 16;

    const int* src = ei;         // edge_index[0]
    const int* dst = ei + E;     // edge_index[1]

    // Workspace layout (fp32): dinv[N] | hbuf[N*16] | agg[N*16] | pool[G*16] | cnt[G]
    float* ws   = (float*)d_ws;
    size_t o    = 0;
    float* dinv = ws + o; o += ((size_t)N + 3) & ~(size_t)3;
    float* hbuf = ws + o; o += (size_t)N * 16;
    float* agg  = ws + o; o += (size_t)N * 16;
    float* pool = ws + o; o += (size_t)G * 16;
    float* cnt  = ws + o;                      // contiguous after pool

    const int B = 256;
    auto nb = [](long n, int b) { return (int)((n + b - 1) / b); };
    const int ntiles = (N + 15) >> 4;
    const int gemm_blocks = nb(ntiles, 8);     // 8 waves per 256-thread block

    // --- normalization ---
    fill_f32<<<nb(N, B), B, 0, stream>>>(dinv, 1.0f, N);                 // self-loop deg
    fill_f32<<<nb((long)N * 16, B), B, 0, stream>>>(agg, 0.0f, N * 16);
    fill_f32<<<nb((long)G * 17, B), B, 0, stream>>>(pool, 0.0f, G * 17); // pool + cnt
    deg_accum<<<nb(E, B), B, 0, stream>>>(dst, dinv, E);
    rsqrt_inplace<<<nb(N, B), B, 0, stream>>>(dinv, N);

    // --- layer 1: h1 = X @ W1 (WMMA), aggregate, +b1, ReLU ---
    gcn_gemm_wmma<5><<<gemm_blocks, B, 0, stream>>>(x, W1, hbuf, N);
    edge_scatter<<<nb(E, B), B, 0, stream>>>(hbuf, src, dst, dinv, agg, E);
    bias_relu_self<<<nb(N, B), B, 0, stream>>>(agg, hbuf, dinv, b1, N);  // agg := h1'

    // --- layer 2: h2 = h1' @ W2 (WMMA), aggregate, +b2, ReLU fused with pooling ---
    gcn_gemm_wmma<16><<<gemm_blocks, B, 0, stream>>>(agg, W2, hbuf, N);  // hbuf := h2
    fill_f32<<<nb((long)N * 16, B), B, 0, stream>>>(agg, 0.0f, N * 16);
    edge_scatter<<<nb(E, B), B, 0, stream>>>(hbuf, src, dst, dinv, agg, E);
    relu_self_pool<<<nb(N, B), B, 0, stream>>>(agg, hbuf, dinv, b2, batch, pool, cnt, N);

    // --- global mean pool ---
    finalize_mean<<<nb((long)G * 16, B), B, 0, stream>>>(pool, cnt, out, G * 16);
}